// GCN_42073499631701
// MI455X (gfx1250) — compile-verified
//
#include <hip/hip_runtime.h>

typedef float v2f __attribute__((ext_vector_type(2)));
typedef float v8f __attribute__((ext_vector_type(8)));

#define CH 128

// ---------------------------------------------------------------------------
// init: agg1 = 0, deg = 1.0 (self loop), out = broadcast(b2)
// ---------------------------------------------------------------------------
__global__ void gcn_init_kernel(float* __restrict__ deg, float* __restrict__ agg1,
                                float* __restrict__ out, const float* __restrict__ b2,
                                int n) {
    size_t i = (size_t)blockIdx.x * blockDim.x + threadIdx.x;
    size_t total = (size_t)n * CH;
    if (i < total)        agg1[i] = 0.0f;
    if (i < (size_t)n)    deg[i]  = 1.0f;          // self-loop contribution
    if (i < 2 * (size_t)n) out[i] = b2[i & 1];     // fold b2 into accumulator init
}

// ---------------------------------------------------------------------------
// deg[dst] += 1 per edge
// ---------------------------------------------------------------------------
__global__ void gcn_deg_kernel(const long long* __restrict__ dst, float* deg, int E) {
    int e = blockIdx.x * blockDim.x + threadIdx.x;
    if (e < E) atomicAdd(&deg[(int)dst[e]], 1.0f);
}

// ---------------------------------------------------------------------------
// deg -> deg^{-1/2} in place (deg >= 1 always due to self loops)
// ---------------------------------------------------------------------------
__global__ void gcn_rsqrt_kernel(float* deg, int n) {
    int i = blockIdx.x * blockDim.x + threadIdx.x;
    if (i < n) {
        float d = deg[i];
        deg[i] = (d > 0.0f) ? rsqrtf(d) : 0.0f;
    }
}

// ---------------------------------------------------------------------------
// h = x @ W1  via V_WMMA_F32_16X16X4_F32.
// Block = 256 threads = 8 waves; wave w owns N-tile [16w,16w+16);
// blockIdx.x owns M-tile.  K looped in steps of 4 (32 WMMAs per wave).
// A 16x4 layout: lanes 0-15 -> M rows (K 0,1 in v[0],v[1]); lanes 16-31 -> K 2,3.
// B 4x16 layout mirrored over N.  C/D: VGPR r holds M = r + 8*(lane>=16), N = lane&15.
// ---------------------------------------------------------------------------
__global__ void gcn_gemm1_wmma(const float* __restrict__ x,
                               const float* __restrict__ W,
                               float* __restrict__ h) {
    const int lane = threadIdx.x & 31;
    const int wave = threadIdx.x >> 5;
    const int m0   = blockIdx.x * 16;
    const int n0   = wave * 16;
    const int half = lane >> 4;     // 0: K={0,1}, 1: K={2,3}
    const int mr   = lane & 15;

    const float* __restrict__ arow = x + (size_t)(m0 + mr) * CH + 2 * half;
    const float* __restrict__ bcol = W + (size_t)(2 * half) * CH + (n0 + mr);

    v8f acc = {};
#pragma unroll
    for (int kb = 0; kb < CH; kb += 4) {
        v2f a = *(const v2f*)(arow + kb);          // K = kb+2h, kb+2h+1  (8B aligned)
        v2f b;
        b.x = bcol[(size_t)kb * CH];               // K = kb+2h   row of W
        b.y = bcol[(size_t)(kb + 1) * CH];         // K = kb+2h+1 row of W
        acc = __builtin_amdgcn_wmma_f32_16x16x4_f32(
            /*neg_a=*/false, a, /*neg_b=*/false, b,
            /*c_mod=*/(short)0, acc, /*reuse_a=*/false, /*reuse_b=*/false);
    }

#pragma unroll
    for (int r = 0; r < 8; ++r) {
        h[(size_t)(m0 + r + 8 * half) * CH + (n0 + mr)] = acc[r];
    }
}

// ---------------------------------------------------------------------------
// Layer-1 scatter: one wave per (edge or self-loop); lane owns 4 channels.
// agg1[d] += dinv[s]*dinv[d] * h[s]    (float atomics, L2-resident)
// ---------------------------------------------------------------------------
__global__ void gcn_scatter1_kernel(const long long* __restrict__ src,
                                    const long long* __restrict__ dst,
                                    const float* __restrict__ dinv,
                                    const float* __restrict__ h,
                                    float* __restrict__ agg1,
                                    int E, int n) {
    long long tid = (long long)blockIdx.x * blockDim.x + threadIdx.x;
    long long e   = tid >> 5;
    int       q   = (int)(tid & 31);
    if (e >= (long long)E + n) return;

    int s, d;
    if (e < E) { s = (int)src[e]; d = (int)dst[e]; }
    else       { s = d = (int)(e - E); }            // self loop

    float norm = dinv[s] * dinv[d];
    const float4 hv = *(const float4*)(h + (size_t)s * CH + q * 4);
    float* ap = agg1 + (size_t)d * CH + q * 4;
    atomicAdd(ap + 0, norm * hv.x);
    atomicAdd(ap + 1, norm * hv.y);
    atomicAdd(ap + 2, norm * hv.z);
    atomicAdd(ap + 3, norm * hv.w);
}

// ---------------------------------------------------------------------------
// h2 = relu(agg1 + b1) @ W2   (128 -> 2), thread per node.
// ---------------------------------------------------------------------------
__global__ void gcn_layer2_kernel(const float* __restrict__ agg1,
                                  const float* __restrict__ b1,
                                  const float* __restrict__ W2,
                                  float* __restrict__ h2, int n) {
    int i = blockIdx.x * blockDim.x + threadIdx.x;
    if (i >= n) return;
    const float* row = agg1 + (size_t)i * CH;
    float a0 = 0.0f, a1 = 0.0f;
#pragma unroll 8
    for (int c = 0; c < CH; ++c) {
        float v = row[c] + b1[c];
        v = v > 0.0f ? v : 0.0f;
        a0 = fmaf(v, W2[2 * c + 0], a0);
        a1 = fmaf(v, W2[2 * c + 1], a1);
    }
    h2[2 * (size_t)i + 0] = a0;
    h2[2 * (size_t)i + 1] = a1;
}

// ---------------------------------------------------------------------------
// Layer-2 scatter: thread per (edge or self-loop), 2 output channels.
// out was pre-initialized with b2.
// ---------------------------------------------------------------------------
__global__ void gcn_scatter2_kernel(const long long* __restrict__ src,
                                    const long long* __restrict__ dst,
                                    const float* __restrict__ dinv,
                                    const float* __restrict__ h2,
                                    float* __restrict__ out,
                                    int E, int n) {
    long long e = (long long)blockIdx.x * blockDim.x + threadIdx.x;
    if (e >= (long long)E + n) return;
    int s, d;
    if (e < E) { s = (int)src[e]; d = (int)dst[e]; }
    else       { s = d = (int)(e - E); }
    float norm = dinv[s] * dinv[d];
    atomicAdd(out + 2 * (size_t)d + 0, norm * h2[2 * (size_t)s + 0]);
    atomicAdd(out + 2 * (size_t)d + 1, norm * h2[2 * (size_t)s + 1]);
}

// ---------------------------------------------------------------------------
extern "C" void kernel_launch(void* const* d_in, const int* in_sizes, int n_in,
                              void* d_out, int out_size, void* d_ws, size_t ws_size,
                              hipStream_t stream) {
    const float*     x   = (const float*)d_in[0];
    const long long* ei  = (const long long*)d_in[1];   // int64 per reference
    const float*     W1  = (const float*)d_in[2];
    const float*     b1  = (const float*)d_in[3];
    const float*     W2  = (const float*)d_in[4];
    const float*     b2  = (const float*)d_in[5];
    float*           out = (float*)d_out;

    const int n = in_sizes[0] / CH;       // 50000
    const int E = in_sizes[1] / 2;        // 800000
    const long long* src = ei;
    const long long* dst = ei + E;

    // Workspace layout (floats), 16B-aligned sections
    float* ws   = (float*)d_ws;
    size_t nAl  = ((size_t)n + 31) & ~(size_t)31;
    float* dinv = ws;                                   // n
    float* h    = dinv + nAl;                           // n*128
    float* agg1 = h + (size_t)n * CH;                   // n*128
    float* h2   = agg1 + (size_t)n * CH;                // 2n

    // 1) init accumulators / degrees / output(+bias2)
    {
        size_t total = (size_t)n * CH;
        int blocks = (int)((total + 255) / 256);
        gcn_init_kernel<<<blocks, 256, 0, stream>>>(dinv, agg1, out, b2, n);
    }
    // 2) degrees over dst (+ self loop already in init)
    gcn_deg_kernel<<<(E + 255) / 256, 256, 0, stream>>>(dst, dinv, E);
    // 3) deg^{-1/2}
    gcn_rsqrt_kernel<<<(n + 255) / 256, 256, 0, stream>>>(dinv, n);
    // 4) h = x @ W1 (WMMA; n = 50000 is a multiple of 16)
    gcn_gemm1_wmma<<<n / 16, 256, 0, stream>>>(x, W1, h);
    // 5) layer-1 normalized scatter-add (wave per edge)
    {
        size_t total = ((size_t)E + n) * 32;
        int blocks = (int)((total + 255) / 256);
        gcn_scatter1_kernel<<<blocks, 256, 0, stream>>>(src, dst, dinv, h, agg1, E, n);
    }
    // 6) relu(agg1 + b1) @ W2
    gcn_layer2_kernel<<<(n + 255) / 256, 256, 0, stream>>>(agg1, b1, W2, h2, n);
    // 7) layer-2 scatter into out (bias already there)
    {
        int total = E + n;
        gcn_scatter2_kernel<<<(total + 255) / 256, 256, 0, stream>>>(src, dst, dinv, h2, out, E, n);
    }
}